// RGAT_81028853006652
// MI455X (gfx1250) — compile-verified
//
#include <hip/hip_runtime.h>
#include <hip/hip_bf16.h>
#include <math.h>

typedef float v2f __attribute__((ext_vector_type(2)));
typedef float v8f __attribute__((ext_vector_type(8)));

#define NNODES 20000
#define NEDGES 400000
#define NREL   8
#define NEG_SLOPE 0.2f

// ---------------------------------------------------------------------------
// Tiled WMMA GEMM (fp32, v_wmma_f32_16x16x4_f32).
// C_z[m, n] = sum_k A[m,k] * B_z[k,n]
//   A: [M x K] row-major.
//   B_z = B + z*batchStrideB, [K x Nc] row-major.
//   C_z = C + z*colOffC, row stride ldc.
// Block: 128 threads (4 waves). Block tile: 64(M) x 64(N). K slab: 16.
// Each wave owns a 16-row strip and four 16x16 accumulators.
// ---------------------------------------------------------------------------
__global__ __launch_bounds__(128)
void wmma_gemm_kernel(const float* __restrict__ A, const float* __restrict__ B,
                      float* __restrict__ C, int M, int K, int Nc,
                      int ldc, int batchStrideB, int colOffC)
{
    const int z = blockIdx.z;
    const float* Bz = B + (size_t)z * batchStrideB;
    float* Cz = C + (size_t)z * colOffC;

    // padded so that float4 stores stay 16B aligned (row strides multiple of 4)
    __shared__ __align__(16) float As[64][20];  // [row][k]
    __shared__ __align__(16) float Bs[16][68];  // [k][col]

    const int tid  = threadIdx.x;      // 0..127
    const int wave = tid >> 5;         // 0..3
    const int lane = tid & 31;
    const int half = lane >> 4;        // 0: lanes 0-15, 1: lanes 16-31
    const int lm   = lane & 15;

    const int m0 = blockIdx.x * 64;
    const int n0 = blockIdx.y * 64;

    v8f acc[4];
#pragma unroll
    for (int t = 0; t < 4; ++t)
#pragma unroll
        for (int i = 0; i < 8; ++i) acc[t][i] = 0.0f;

    for (int k0 = 0; k0 < K; k0 += 16) {
        // ---- stage A slab: 64 rows x 16 k (128 thr x 8 elem) ----
        {
            int idx = tid * 8;
            int row = idx >> 4;        // 0..63
            int kk  = idx & 15;        // 0 or 8
            int gm  = m0 + row;
            float4 v0, v1;
            if (gm < M) {
                const float* s = A + (size_t)gm * K + k0 + kk;
                v0 = *(const float4*)(s);
                v1 = *(const float4*)(s + 4);
                if (k0 + 16 < K) __builtin_prefetch(s + 16); // next slab
            } else {
                v0 = make_float4(0.f, 0.f, 0.f, 0.f); v1 = v0;
            }
            *(float4*)(&As[row][kk])     = v0;
            *(float4*)(&As[row][kk + 4]) = v1;
        }
        // ---- stage B slab: 16 k x 64 cols ----
        {
            int idx = tid * 8;
            int kr  = idx >> 6;        // 0..15
            int nn  = idx & 63;        // multiple of 8
            const float* s = Bz + (size_t)(k0 + kr) * Nc + n0 + nn;
            *(float4*)(&Bs[kr][nn])     = *(const float4*)(s);
            *(float4*)(&Bs[kr][nn + 4]) = *(const float4*)(s + 4);
        }
        __syncthreads();

        const int wr = wave * 16;
#pragma unroll
        for (int kk = 0; kk < 16; kk += 4) {
            // A frag 16x4: VGPR0 = K0(l0-15)/K2(l16-31), VGPR1 = K1/K3
            v2f a;
            a.x = As[wr + lm][kk + half * 2 + 0];
            a.y = As[wr + lm][kk + half * 2 + 1];
#pragma unroll
            for (int nt = 0; nt < 4; ++nt) {
                // B frag 4x16: VGPR0 = row kb (cols striped over lanes), VGPR1 = row kb+1
                v2f b;
                b.x = Bs[kk + half * 2 + 0][nt * 16 + lm];
                b.y = Bs[kk + half * 2 + 1][nt * 16 + lm];
                acc[nt] = __builtin_amdgcn_wmma_f32_16x16x4_f32(
                    false, a, false, b, (short)0, acc[nt], false, false);
            }
        }
        __syncthreads();
    }

    // ---- store: C VGPR i -> row i (lanes 0-15) / row i+8 (lanes 16-31) ----
#pragma unroll
    for (int nt = 0; nt < 4; ++nt) {
        int col = n0 + nt * 16 + lm;
#pragma unroll
        for (int i = 0; i < 8; ++i) {
            int row = m0 + wave * 16 + half * 8 + i;
            if (row < M) Cz[(size_t)row * ldc + col] = acc[nt][i];
        }
    }
}

// ---------------------------------------------------------------------------
// rel_proj[r][j] = rel_emb[r] @ rel_proj_w[:,j] + b[j]   (8x256, tiny)
// ---------------------------------------------------------------------------
__global__ void rel_proj_kernel(const float* __restrict__ rel_emb,
                                const float* __restrict__ w,
                                const float* __restrict__ b,
                                float* __restrict__ outp)
{
    int r = blockIdx.x, j = threadIdx.x;   // 8 blocks x 256 threads
    float acc = b[j];
    for (int k = 0; k < 256; ++k) acc += rel_emb[r * 256 + k] * w[k * 256 + j];
    outp[r * 256 + j] = acc;
}

// per-(dst,relation) edge count histogram (float so the GEMM is direct)
__global__ void count_kernel(const int* __restrict__ dst,
                             const int* __restrict__ etype,
                             float* __restrict__ cnt)
{
    int e = blockIdx.x * blockDim.x + threadIdx.x;
    if (e >= NEDGES) return;
    atomicAdd(&cnt[(size_t)dst[e] * NREL + etype[e]], 1.0f);
}

// rel_agg = cnt[N,8] @ rel_proj[8,256]
__global__ void rel_agg_kernel(const float* __restrict__ cnt,
                               const float* __restrict__ rel_proj,
                               float* __restrict__ rel_agg)
{
    int n = blockIdx.x, j = threadIdx.x;
    float acc = 0.f;
#pragma unroll
    for (int r = 0; r < NREL; ++r)
        acc += cnt[(size_t)n * NREL + r] * rel_proj[r * 256 + j];
    rel_agg[(size_t)n * 256 + j] = acc;
}

// s_dst[n] = root[n] . att_dst   (one wave32 per node)
__global__ void sdst_kernel(const float* __restrict__ root,
                            const float* __restrict__ att_dst,
                            float* __restrict__ sdst, int do_)
{
    int wid = blockIdx.x * (blockDim.x >> 5) + (threadIdx.x >> 5);
    if (wid >= NNODES) return;
    int lane = threadIdx.x & 31;
    const float* row = root + (size_t)wid * do_;
    float acc = 0.f;
    for (int j = lane; j < do_; j += 32) acc += row[j] * att_dst[j];
#pragma unroll
    for (int off = 16; off > 0; off >>= 1) acc += __shfl_xor(acc, off, 32);
    if (lane == 0) sdst[wid] = acc;
}

__device__ __forceinline__ unsigned float_to_ordered(float f) {
    unsigned u = __float_as_uint(f);
    return (u & 0x80000000u) ? ~u : (u | 0x80000000u);
}
__device__ __forceinline__ float ordered_to_float(unsigned k) {
    unsigned u = (k & 0x80000000u) ? (k ^ 0x80000000u) : ~k;
    return __uint_as_float(u);
}

// score[e] = leaky_relu(msg_e . att_src + s_dst[dst]); segment-max via atomicMax
__global__ void edge_score_kernel(const int* __restrict__ src,
                                  const int* __restrict__ dst,
                                  const int* __restrict__ etype,
                                  const float* __restrict__ hr,
                                  const float* __restrict__ att_src,
                                  const float* __restrict__ sdst,
                                  float* __restrict__ score,
                                  unsigned* __restrict__ mkey, int do_)
{
    int e = blockIdx.x * (blockDim.x >> 5) + (threadIdx.x >> 5);
    if (e >= NEDGES) return;
    int lane = threadIdx.x & 31;
    int s = src[e], d = dst[e], t = etype[e];
    const float* msg = hr + ((size_t)s * NREL + t) * do_;
    float acc = 0.f;
    for (int j = lane; j < do_; j += 32) acc += msg[j] * att_src[j];
#pragma unroll
    for (int off = 16; off > 0; off >>= 1) acc += __shfl_xor(acc, off, 32);
    if (lane == 0) {
        float sc = acc + sdst[d];
        sc = (sc > 0.f) ? sc : NEG_SLOPE * sc;
        score[e] = sc;
        atomicMax(&mkey[d], float_to_ordered(sc)); // memset-0 is identity
    }
}

// ex[e] = exp(score - m[dst]); denom[dst] += ex
__global__ void edge_exp_kernel(const int* __restrict__ dst,
                                const float* __restrict__ score,
                                const unsigned* __restrict__ mkey,
                                float* __restrict__ exb,
                                float* __restrict__ denom)
{
    int e = blockIdx.x * blockDim.x + threadIdx.x;
    if (e >= NEDGES) return;
    int d = dst[e];
    float m = ordered_to_float(mkey[d]);
    float ex = __expf(score[e] - m);
    exb[e] = ex;
    atomicAdd(&denom[d], ex);
}

// agg[dst] += (ex/denom[dst]) * msg_e   (one wave32 per edge)
__global__ void edge_agg_kernel(const int* __restrict__ src,
                                const int* __restrict__ dst,
                                const int* __restrict__ etype,
                                const float* __restrict__ hr,
                                const float* __restrict__ exb,
                                const float* __restrict__ denom,
                                float* __restrict__ agg, int do_)
{
    int e = blockIdx.x * (blockDim.x >> 5) + (threadIdx.x >> 5);
    if (e >= NEDGES) return;
    int lane = threadIdx.x & 31;
    int s = src[e], d = dst[e], t = etype[e];
    float alpha = exb[e] / denom[d];
    const float* msg = hr + ((size_t)s * NREL + t) * do_;
    float* out = agg + (size_t)d * do_;
    for (int j = lane; j < do_; j += 32) atomicAdd(&out[j], alpha * msg[j]);
}

// hidden layers: h_next = h + relu(root + agg + rel_agg)
__global__ void finalize_hidden(const float* __restrict__ h,
                                const float* __restrict__ root,
                                const float* __restrict__ agg,
                                const float* __restrict__ rel_agg,
                                float* __restrict__ hnext, int total)
{
    int i = blockIdx.x * blockDim.x + threadIdx.x;
    if (i >= total) return;
    float v = root[i] + agg[i] + rel_agg[i];
    v = (v > 0.f) ? v : 0.f;
    hnext[i] = h[i] + v;
}

// final layer: out = root + agg
__global__ void finalize_out(const float* __restrict__ root,
                             const float* __restrict__ agg,
                             float* __restrict__ outp, int total)
{
    int i = blockIdx.x * blockDim.x + threadIdx.x;
    if (i >= total) return;
    outp[i] = root[i] + agg[i];
}

// ---------------------------------------------------------------------------
extern "C" void kernel_launch(void* const* d_in, const int* in_sizes, int n_in,
                              void* d_out, int out_size, void* d_ws, size_t ws_size,
                              hipStream_t stream)
{
    (void)in_sizes; (void)n_in; (void)out_size; (void)ws_size;

    const float* x          = (const float*)d_in[0];
    const int*   ei         = (const int*)  d_in[1];
    const int*   etype      = (const int*)  d_in[2];
    const float* rel_emb    = (const float*)d_in[3];
    const float* rel_proj_w = (const float*)d_in[4];
    const float* rel_proj_b = (const float*)d_in[5];
    const float* W_rel[3]   = {(const float*)d_in[6],  (const float*)d_in[10], (const float*)d_in[14]};
    const float* W_root[3]  = {(const float*)d_in[7],  (const float*)d_in[11], (const float*)d_in[15]};
    const float* att_src[3] = {(const float*)d_in[8],  (const float*)d_in[12], (const float*)d_in[16]};
    const float* att_dst[3] = {(const float*)d_in[9],  (const float*)d_in[13], (const float*)d_in[17]};
    const int* src = ei;            // edge_index[0]
    const int* dst = ei + NEDGES;   // edge_index[1]

    // workspace layout (floats)
    float* ws = (float*)d_ws;
    size_t off = 0;
    auto alloc = [&](size_t n) { float* p = ws + off; off += n; return p; };
    float*    rel_proj = alloc((size_t)NREL * 256);
    float*    cnt      = alloc((size_t)NNODES * NREL);
    float*    rel_agg  = alloc((size_t)NNODES * 256);
    float*    hA       = alloc((size_t)NNODES * 256);
    float*    hB       = alloc((size_t)NNODES * 256);
    float*    hr       = alloc((size_t)NNODES * NREL * 256);
    float*    root     = alloc((size_t)NNODES * 256);
    float*    sdst     = alloc((size_t)NNODES);
    float*    score    = alloc((size_t)NEDGES);
    float*    exb      = alloc((size_t)NEDGES);
    float*    denom    = alloc((size_t)NNODES);
    float*    agg      = alloc((size_t)NNODES * 256);
    unsigned* mkey     = (unsigned*)alloc((size_t)NNODES);

    // ---- relation-embedding aggregation: 8 distinct rows -> histogram GEMM ----
    hipMemsetAsync(cnt, 0, (size_t)NNODES * NREL * sizeof(float), stream);
    rel_proj_kernel<<<NREL, 256, 0, stream>>>(rel_emb, rel_proj_w, rel_proj_b, rel_proj);
    count_kernel<<<(NEDGES + 255) / 256, 256, 0, stream>>>(dst, etype, cnt);
    rel_agg_kernel<<<NNODES, 256, 0, stream>>>(cnt, rel_proj, rel_agg);

    const int din[3]  = {256, 256, 256};
    const int dout[3] = {256, 256, 128};
    const float* h = x;
    float* hnext[2] = {hA, hB};

    for (int l = 0; l < 3; ++l) {
        const int di = din[l], dO = dout[l];

        hipMemsetAsync(mkey,  0, (size_t)NNODES * sizeof(unsigned), stream);
        hipMemsetAsync(denom, 0, (size_t)NNODES * sizeof(float), stream);
        hipMemsetAsync(agg,   0, (size_t)NNODES * dO * sizeof(float), stream);

        // hr[n, r, :] = h @ W_rel[r]   (batched over r via grid.z)
        dim3 g1((NNODES + 63) / 64, dO / 64, NREL);
        wmma_gemm_kernel<<<g1, 128, 0, stream>>>(h, W_rel[l], hr,
                                                 NNODES, di, dO,
                                                 NREL * dO, di * dO, dO);
        // root = h @ W_root
        dim3 g2((NNODES + 63) / 64, dO / 64, 1);
        wmma_gemm_kernel<<<g2, 128, 0, stream>>>(h, W_root[l], root,
                                                 NNODES, di, dO, dO, 0, 0);

        sdst_kernel<<<(NNODES + 7) / 8, 256, 0, stream>>>(root, att_dst[l], sdst, dO);
        edge_score_kernel<<<(NEDGES + 7) / 8, 256, 0, stream>>>(src, dst, etype, hr,
                                                                att_src[l], sdst,
                                                                score, mkey, dO);
        edge_exp_kernel<<<(NEDGES + 255) / 256, 256, 0, stream>>>(dst, score, mkey, exb, denom);
        edge_agg_kernel<<<(NEDGES + 7) / 8, 256, 0, stream>>>(src, dst, etype, hr,
                                                              exb, denom, agg, dO);

        int total = NNODES * dO;
        if (l < 2) {
            finalize_hidden<<<(total + 255) / 256, 256, 0, stream>>>(h, root, agg, rel_agg,
                                                                     hnext[l], total);
            h = hnext[l];
        } else {
            finalize_out<<<(total + 255) / 256, 256, 0, stream>>>(root, agg, (float*)d_out, total);
        }
    }
}